// CommunityDetector_44899588112467
// MI455X (gfx1250) — compile-verified
//
#include <hip/hip_runtime.h>
#include <cstdint>
#include <math.h>

#ifndef __has_builtin
#define __has_builtin(x) 0
#endif

#if __has_builtin(__builtin_amdgcn_global_load_async_to_lds_b128)
#define HAVE_ASYNC_LDS 1
#else
#define HAVE_ASYNC_LDS 0
#endif

#define CC 128            // channels
#define DD 64             // neighbors per node
#define NNODES 8192
#define LSTR 132          // padded LDS row stride (floats) -> conflict-free column reads
#define NEG_SLOPE 0.2f

typedef __attribute__((ext_vector_type(2))) float v2f;
typedef __attribute__((ext_vector_type(8))) float v8f;
typedef __attribute__((ext_vector_type(4))) int   v4i;

// pointer-to-addrspace typedefs for the async copy builtin
typedef __attribute__((address_space(1))) v4i* gas_v4i_ptr;   // global
typedef __attribute__((address_space(3))) v4i* las_v4i_ptr;   // LDS

// ---------------------------------------------------------------------------
// K0: once-per-launch precompute.  s_center(node) = max_feature . v + c0 where
//     v[k] = sum_c W_max[k,c] * W_score[c,0],  c0 = b_max . Ws_top + b_score.
// ---------------------------------------------------------------------------
__global__ __launch_bounds__(128) void prep_kernel(
    const float* __restrict__ W_max, const float* __restrict__ b_max,
    const float* __restrict__ W_score, const float* __restrict__ b_score,
    float* __restrict__ ws) {
  int k = threadIdx.x;  // 0..127
  float acc = 0.f;
  for (int c = 0; c < CC; ++c) acc = fmaf(W_max[k * CC + c], W_score[c], acc);
  ws[k] = acc;
  if (k == 0) {
    float c0 = b_score[0];
    for (int c = 0; c < CC; ++c) c0 = fmaf(b_max[c], W_score[c], c0);
    ws[CC] = c0;
  }
}

// ---------------------------------------------------------------------------
// Fused per-node kernel: one block = one source node (64 nbrs x 128 ch in LDS)
// ---------------------------------------------------------------------------
__global__ __launch_bounds__(256) void community_fused_kernel(
    const float* __restrict__ nf, const float* __restrict__ memv,
    const float* __restrict__ W_score, const float* __restrict__ W_fit,
    const float* __restrict__ b_fit, const int* __restrict__ neighbors,
    const float* __restrict__ ws, float* __restrict__ out) {
  __shared__ float nbr[DD * LSTR];      // 64 x 132 (padded) neighbor tile
  __shared__ int   nid[DD];
  __shared__ float vv[CC];              // precomputed v
  __shared__ float wsb[CC];             // W_score rows 128..255
  __shared__ float wfit[CC];
  __shared__ float red[2 * CC];         // 2-half reduction scratch
  __shared__ float maxf[CC];
  __shared__ float partial_s[2][DD];    // per-k-half WMMA partial dot products
  __shared__ float sarr[DD];
  __shared__ float score[DD];
  __shared__ float cf[CC];
  __shared__ float sc_center, c0_s;

  const int t = threadIdx.x;
  const int n = blockIdx.x;

  // ---- phase 0: stage ids + small constants ------------------------------
  if (t < DD) nid[t] = neighbors[n * DD + t];
  if (t >= 64 && t < 192) {
    int c = t - 64;
    vv[c]   = ws[c];
    wsb[c]  = W_score[CC + c];
    wfit[c] = W_fit[c];
  }
  if (t == 0) c0_s = ws[CC];
  __syncthreads();

  // ---- phase 1: gather nbr = feat + mem into LDS -------------------------
  // 64 rows x 32 float4 chunks = 2048 chunks; 8 per thread.
  float4 mval[8];
#if !HAVE_ASYNC_LDS
  float4 fval[8];
#endif
  for (int i = 0; i < 8; ++i) {
    int q = t + 256 * i;
    int j = q >> 5, k = q & 31;
    long off = (long)nid[j] * CC + (long)(k * 4);
#if HAVE_ASYNC_LDS
    __builtin_amdgcn_global_load_async_to_lds_b128(
        (gas_v4i_ptr)(nf + off),
        (las_v4i_ptr)(&nbr[j * LSTR + k * 4]),
        0, 0);
#else
    fval[i] = *(const float4*)(nf + off);
#endif
    mval[i] = *(const float4*)(memv + off);
  }
#if HAVE_ASYNC_LDS
#if __has_builtin(__builtin_amdgcn_s_wait_asynccnt)
  __builtin_amdgcn_s_wait_asynccnt(0);
#else
  asm volatile("s_wait_asynccnt 0" ::: "memory");
#endif
  asm volatile("" ::: "memory");  // compiler barrier: LDS now holds async data
#endif
  for (int i = 0; i < 8; ++i) {
    int q = t + 256 * i;
    int j = q >> 5, k = q & 31;
    float4* p = (float4*)&nbr[j * LSTR + k * 4];
#if HAVE_ASYNC_LDS
    float4 x = *p;
#else
    float4 x = fval[i];
#endif
    x.x += mval[i].x; x.y += mval[i].y; x.z += mval[i].z; x.w += mval[i].w;
    *p = x;
  }
  __syncthreads();

  // ---- phase 2: segment max over neighbors (per channel) -----------------
  {
    int c = t & (CC - 1), half = t >> 7;
    float m = -INFINITY;
    for (int j = half * 32; j < half * 32 + 32; ++j)
      m = fmaxf(m, nbr[j * LSTR + c]);
    red[half * CC + c] = m;
  }
  __syncthreads();
  if (t < CC) maxf[t] = fmaxf(red[t], red[CC + t]);
  __syncthreads();

  // ---- phase 3: per-node center score (wave 0) ---------------------------
  if (t < 32) {
    float p = 0.f;
    for (int i = 0; i < 4; ++i) {
      int c = t + 32 * i;
      p = fmaf(maxf[c], vv[c], p);
    }
    for (int o = 16; o > 0; o >>= 1) p += __shfl_xor(p, o, 32);
    if (t == 0) sc_center = p + c0_s;
  }

  // ---- phase 4: neighbor dot products via V_WMMA_F32_16X16X4_F32 ---------
  // wave w: row-group g = w&3 (16 rows), k-half h = w>>2 (64 channels, 16
  // K-chunks of 4).  B[k][*] = wsb[k] replicated -> every D column = s_raw.
  {
    const int lane = t & 31;
    const int g = (t >> 5) & 3;
    const int h = t >> 7;
    const int m = lane & 15;
    const int kk = (lane >> 4) * 2;     // A layout: lanes<16 -> K0,K1; else K2,K3
    const float* nrow = &nbr[(g * 16 + m) * LSTR];
    v8f acc = {};
    for (int kc = 0; kc < 16; ++kc) {
      int k0 = (h * 16 + kc) * 4;
      v2f a = *(const v2f*)&nrow[k0 + kk];
      v2f b = *(const v2f*)&wsb[k0 + kk];
      acc = __builtin_amdgcn_wmma_f32_16x16x4_f32(
          false, a, false, b, (short)0, acc, false, false);
    }
    if (m == 0) {  // lanes 0 and 16: D vgpr r holds rows r / r+8, all cols equal
      int rbase = g * 16 + (lane >> 4) * 8;
      for (int r = 0; r < 8; ++r) partial_s[h][rbase + r] = acc[r];
    }
  }
  __syncthreads();

  // ---- phase 5: leaky-relu score, then segment softmax -------------------
  if (t < DD) {
    float s = partial_s[0][t] + partial_s[1][t] + sc_center;
    sarr[t] = (s >= 0.f) ? s : NEG_SLOPE * s;
  }
  __syncthreads();
  if (t < 32) {
    float s0 = sarr[t], s1 = sarr[t + 32];
    float mm = fmaxf(s0, s1);
    for (int o = 16; o > 0; o >>= 1) mm = fmaxf(mm, __shfl_xor(mm, o, 32));
    float e0 = __expf(s0 - mm), e1 = __expf(s1 - mm);
    float ssum = e0 + e1;
    for (int o = 16; o > 0; o >>= 1) ssum += __shfl_xor(ssum, o, 32);
    score[t] = e0 / ssum;
    score[t + 32] = e1 / ssum;
  }
  __syncthreads();

  // ---- phase 6: cluster_feature = sum_j score[j] * nbr[j][:] -------------
  {
    int c = t & (CC - 1), half = t >> 7;
    float a = 0.f;
    for (int j = half * 32; j < half * 32 + 32; ++j)
      a = fmaf(score[j], nbr[j * LSTR + c], a);
    red[half * CC + c] = a;
  }
  __syncthreads();
  if (t < CC) {
    float v = red[t] + red[CC + t];
    cf[t] = v;
    out[(long)n * CC + t] = v;                       // cluster_feature
  }
  if (t >= CC && t < CC + DD)
    out[(long)NNODES * CC + (long)n * DD + (t - CC)] = score[t - CC];  // score_tensor
  __syncthreads();

  // ---- phase 7: fitness = sigmoid(cf . W_fit + b_fit) (wave 0) -----------
  if (t < 32) {
    float p = 0.f;
    for (int i = 0; i < 4; ++i) {
      int c = t + 32 * i;
      p = fmaf(cf[c], wfit[c], p);
    }
    for (int o = 16; o > 0; o >>= 1) p += __shfl_xor(p, o, 32);
    if (t == 0) {
      float x = p + b_fit[0];
      out[(long)NNODES * CC + (long)NNODES * DD + n] = 1.0f / (1.0f + __expf(-x));
    }
  }
}

// ---------------------------------------------------------------------------
extern "C" void kernel_launch(void* const* d_in, const int* in_sizes, int n_in,
                              void* d_out, int out_size, void* d_ws, size_t ws_size,
                              hipStream_t stream) {
  const float* nf       = (const float*)d_in[0];  // node_features [200000,128]
  const float* memr     = (const float*)d_in[1];  // memory        [200000,128]
  const float* W_max    = (const float*)d_in[2];  // [128,128]
  const float* b_max    = (const float*)d_in[3];  // [128]
  const float* W_score  = (const float*)d_in[4];  // [256,1]
  const float* b_score  = (const float*)d_in[5];  // [1]
  const float* W_fit    = (const float*)d_in[6];  // [128,1]
  const float* b_fit    = (const float*)d_in[7];  // [1]
  const int*   neighbors= (const int*)  d_in[8];  // [524288]
  // d_in[9] = seg, implicit (repeat(arange(8192), 64))

  float* ws  = (float*)d_ws;    // [0..127]=v, [128]=c0
  float* out = (float*)d_out;   // cf(8192*128) | score(8192*64) | fitness(8192)

  prep_kernel<<<1, 128, 0, stream>>>(W_max, b_max, W_score, b_score, ws);
  community_fused_kernel<<<NNODES, 256, 0, stream>>>(
      nf, memr, W_score, W_fit, b_fit, neighbors, ws, out);
}